// NodeRelationExtractor_51573967290519
// MI455X (gfx1250) — compile-verified
//
#include <hip/hip_runtime.h>
#include <math.h>

// ---------------------------------------------------------------------------
// NodeRelationExtractor, MI455X / gfx1250 (wave32, WMMA)
//
// B=2, N=1024, D=256, H2=128.  out[b,n,m] = g(corr[b,n,m]) where
//   corr = clip( (fc @ sc^T / 16) / max(|fc_n||sc_m|, 1e-6), -1, 1 )
//   g(c)  = b2 + sum_h w2[h] * gelu_exact( exp(c)*w1[h] + b1[h] )
// g is a smooth 1-D function on exactly [-1,1] -> tabulated once (4096
// intervals, exact erf), then each of the 2M outputs is a lerp.
// The GEMM uses V_WMMA_F32_16X16X4_F32 (fp32 in/out, full precision).
// Epilogue uses v_rcp_f32 instead of IEEE divide: quotient is clipped and
// quantized at 4.9e-4 granularity, so 1-ulp rcp error is invisible.
// ---------------------------------------------------------------------------

typedef __attribute__((ext_vector_type(2))) float v2f;
typedef __attribute__((ext_vector_type(8))) float v8f;

#define NB    2
#define NN    1024
#define ND    256
#define NH    128
#define TAB_N 4096   // intervals; table has TAB_N+1 entries

// ---------------------------------------------------------------------------
// Kernel 1: per-row centering + std (sqrt of sum of squares, clamped 1e-6)
// One 256-thread block per row (handles both functional and structural rows).
// ---------------------------------------------------------------------------
__device__ __forceinline__ float block_sum_256(float v, float* red) {
    const int tid = threadIdx.x;
    red[tid] = v;
    __syncthreads();
    #pragma unroll
    for (int s = 128; s > 0; s >>= 1) {
        if (tid < s) red[tid] += red[tid + s];
        __syncthreads();
    }
    float r = red[0];
    __syncthreads();
    return r;
}

__global__ __launch_bounds__(256) void center_stats_kernel(
    const float* __restrict__ F, const float* __restrict__ S,
    float* __restrict__ fc, float* __restrict__ sc,
    float* __restrict__ std_f, float* __restrict__ std_s)
{
    __shared__ float red[256];
    const int row  = blockIdx.x;              // 0 .. B*N-1
    const int tid  = threadIdx.x;             // 0 .. 255 (== D)
    const size_t base = (size_t)row * ND;

    const float vf = F[base + tid];
    const float vs = S[base + tid];

    const float mf = block_sum_256(vf, red) * (1.0f / (float)ND);
    const float ms = block_sum_256(vs, red) * (1.0f / (float)ND);
    const float cf = vf - mf;
    const float cs = vs - ms;

    const float ssf = block_sum_256(cf * cf, red);
    const float sss = block_sum_256(cs * cs, red);

    fc[base + tid] = cf;
    sc[base + tid] = cs;
    if (tid == 0) {
        std_f[row] = fmaxf(sqrtf(ssf), 1e-6f);
        std_s[row] = fmaxf(sqrtf(sss), 1e-6f);
    }
}

// ---------------------------------------------------------------------------
// Kernel 2: build the 1-D response table  g(c), c uniform on [-1, 1].
// Exact GELU via erff; cost = 4097 * 128 evals (negligible, done once).
// ---------------------------------------------------------------------------
__global__ __launch_bounds__(256) void build_table_kernel(
    const float* __restrict__ w1, const float* __restrict__ b1,
    const float* __restrict__ w2, const float* __restrict__ b2,
    float* __restrict__ table)
{
    const int i = blockIdx.x * blockDim.x + threadIdx.x;
    if (i > TAB_N) return;
    const float corr = -1.0f + (float)i * (2.0f / (float)TAB_N);
    const float a = expf(corr);
    float acc = b2[0];
    #pragma unroll 4
    for (int h = 0; h < NH; ++h) {
        const float x = fmaf(a, w1[h], b1[h]);
        const float g = 0.5f * x * (1.0f + erff(x * 0.70710678118654752440f));
        acc = fmaf(w2[h], g, acc);
    }
    table[i] = acc;
}

// ---------------------------------------------------------------------------
// Kernel 3: fp32 WMMA GEMM (fc @ sc^T) fused with correlation + table lerp.
// One wave per 16x16 output tile; 8 waves (256 threads) per block.
// K-loop: 64 x V_WMMA_F32_16X16X4_F32.
//
// ISA fragment layouts (wave32, 32-bit, ISA 7.12.2):
//   A 16x4 : lane L -> M = L&15;  VGPR0 = A[M][k + 2*(L>>4)], VGPR1 = +1
//   B 4x16 : lane L -> N = L&15;  VGPR0 = B[k + 2*(L>>4)][N], VGPR1 = +1
//            (B = sc^T, so B[k][n] = sc[n][k]: contiguous float2 in sc row)
//   C/D    : elem j -> M = j + 8*(L>>4), N = L&15
// EXEC is all-ones (no divergence around the WMMA; grid divides evenly).
// ---------------------------------------------------------------------------
__global__ __launch_bounds__(256) void cov_mlp_kernel(
    const float* __restrict__ fc, const float* __restrict__ sc,
    const float* __restrict__ std_f, const float* __restrict__ std_s,
    const float* __restrict__ table, float* __restrict__ out)
{
    const int lane = threadIdx.x & 31;
    const int wave = threadIdx.x >> 5;
    const int tile = blockIdx.x * 8 + wave;        // 0 .. 8191
    const int b    = tile >> 12;                   // tile / 4096
    const int tm   = (tile >> 6) & 63;             // row tile
    const int tn   = tile & 63;                    // col tile

    const int r    = lane & 15;
    const int koff = (lane >> 4) << 1;             // 0 or 2

    const float* arow = fc + ((size_t)b * NN + (size_t)tm * 16 + r) * ND + koff;
    const float* brow = sc + ((size_t)b * NN + (size_t)tn * 16 + r) * ND + koff;

    v8f c = {0.f, 0.f, 0.f, 0.f, 0.f, 0.f, 0.f, 0.f};
    #pragma unroll 8
    for (int k = 0; k < ND; k += 4) {
        const v2f av = *(const v2f*)(arow + k);
        const v2f bv = *(const v2f*)(brow + k);
        c = __builtin_amdgcn_wmma_f32_16x16x4_f32(
                /*neg_a=*/false, av, /*neg_b=*/false, bv,
                /*c_mod=*/(short)0, c, /*reuse_a=*/false, /*reuse_b=*/false);
    }

    // Fused epilogue: corr = clip(cov/16 * rcp(max(sf*ss,1e-6))), table lerp.
    // denom >= 1e-6 (clamped), so v_rcp_f32 is safe; its ~1ulp error is far
    // below the table quantization (2/4096).
    const int   col   = tn * 16 + r;
    const float ss    = std_s[b * NN + col];
    const int   row0  = tm * 16 + ((lane >> 4) << 3);   // +8 for upper lanes
    float* orow = out + ((size_t)b << 20) + (size_t)row0 * NN + col;

    #pragma unroll
    for (int j = 0; j < 8; ++j) {
        const float sf    = std_f[b * NN + row0 + j];
        const float denom = fmaxf(sf * ss, 1e-6f);
        const float rden  = __builtin_amdgcn_rcpf(denom);   // v_rcp_f32
        float corr = c[j] * (1.0f / 16.0f) * rden;          // temp = sqrt(256)
        corr = fminf(fmaxf(corr, -1.0f), 1.0f);
        float u  = (corr + 1.0f) * ((float)TAB_N * 0.5f);
        int   i  = (int)u;
        i = (i > TAB_N - 1) ? (TAB_N - 1) : i;
        const float fr = u - (float)i;
        const float t0 = table[i];
        const float t1 = table[i + 1];
        orow[(size_t)j * NN] = fmaf(fr, t1 - t0, t0);
    }
}

// ---------------------------------------------------------------------------
// Launch
// ---------------------------------------------------------------------------
extern "C" void kernel_launch(void* const* d_in, const int* in_sizes, int n_in,
                              void* d_out, int out_size, void* d_ws, size_t ws_size,
                              hipStream_t stream) {
    const float* F  = (const float*)d_in[0];   // [B,N,D]
    const float* S  = (const float*)d_in[1];   // [B,N,D]
    const float* w1 = (const float*)d_in[2];   // [H2]
    const float* b1 = (const float*)d_in[3];   // [H2]
    const float* w2 = (const float*)d_in[4];   // [H2]
    const float* b2 = (const float*)d_in[5];   // [1]
    float* out = (float*)d_out;                // [B,N,N]

    // Workspace layout (floats): fc, sc, std_f, std_s, table (~4.2 MB total)
    float* ws    = (float*)d_ws;
    float* fc    = ws;                                   // B*N*D = 524288
    float* sc    = fc + (size_t)NB * NN * ND;            // 524288
    float* std_f = sc + (size_t)NB * NN * ND;            // 2048
    float* std_s = std_f + (size_t)NB * NN;              // 2048
    float* table = std_s + (size_t)NB * NN;              // 4097

    center_stats_kernel<<<dim3(NB * NN), dim3(256), 0, stream>>>(
        F, S, fc, sc, std_f, std_s);

    build_table_kernel<<<dim3((TAB_N + 1 + 255) / 256), dim3(256), 0, stream>>>(
        w1, b1, w2, b2, table);

    cov_mlp_kernel<<<dim3(NB * 64 * 64 / 8), dim3(256), 0, stream>>>(
        fc, sc, std_f, std_s, table, out);
}